// MessagePassing_23235773072078
// MI455X (gfx1250) — compile-verified
//
#include <hip/hip_runtime.h>
#include <hip/hip_bf16.h>

typedef __attribute__((ext_vector_type(2))) float v2f;
typedef __attribute__((ext_vector_type(8))) float v8f;

// ---------------------------------------------------------------------------
// Pass 1: histogram of destination indices (u32 atomics, cheap)
// ---------------------------------------------------------------------------
__global__ void mp_hist_kernel(const int* __restrict__ dst,
                               unsigned* __restrict__ cnt, int nE) {
    int e = blockIdx.x * blockDim.x + threadIdx.x;
    if (e < nE) {
        atomicAdd(&cnt[dst[e]], 1u);
    }
}

// ---------------------------------------------------------------------------
// Pass 2: exclusive prefix sum over counts (single block, 1024 threads).
// Each thread serially sums a contiguous chunk, block-scans partials in LDS,
// then serially emits exclusive offsets for its chunk.
// ---------------------------------------------------------------------------
__global__ void mp_scan_kernel(const unsigned* __restrict__ cnt,
                               unsigned* __restrict__ ofs,
                               unsigned* __restrict__ cursor, int nNodes) {
    __shared__ unsigned partial[1024];
    const int t = threadIdx.x;
    const int chunk = (nNodes + 1023) / 1024;
    const int lo = t * chunk;
    int hi = lo + chunk;
    if (hi > nNodes) hi = nNodes;

    unsigned s = 0;
    for (int i = lo; i < hi; ++i) s += cnt[i];
    partial[t] = s;
    __syncthreads();

    // Hillis-Steele inclusive scan over 1024 partials
    for (int off = 1; off < 1024; off <<= 1) {
        unsigned v = (t >= off) ? partial[t - off] : 0u;
        __syncthreads();
        partial[t] += v;
        __syncthreads();
    }

    unsigned base = (t == 0) ? 0u : partial[t - 1];  // exclusive base for chunk
    for (int i = lo; i < hi; ++i) {
        ofs[i] = base;
        cursor[i] = base;
        base += cnt[i];
    }
}

// ---------------------------------------------------------------------------
// Pass 3: counting-sort edges by dst into CSR order (u32 slot-claim atomics)
// ---------------------------------------------------------------------------
__global__ void mp_bucket_kernel(const int* __restrict__ src,
                                 const int* __restrict__ dst,
                                 unsigned* __restrict__ cursor,
                                 int* __restrict__ sorted_src, int nE) {
    int e = blockIdx.x * blockDim.x + threadIdx.x;
    if (e < nE) {
        unsigned pos = atomicAdd(&cursor[dst[e]], 1u);
        sorted_src[pos] = src[e];
    }
}

// ---------------------------------------------------------------------------
// Pass 4: atomic-free gather-reduce. 16 lanes per node, one float4 per lane
// (16 lanes x 4 floats = 64 features). Source rows are random but resident in
// the 192MB L2; software-pipeline the next row with global_prefetch.
// Writes RAW segment sums (division happens in the WMMA finalize pass).
// ---------------------------------------------------------------------------
__global__ void mp_reduce_kernel(const float4* __restrict__ x4,
                                 const int* __restrict__ sorted_src,
                                 const unsigned* __restrict__ ofs,
                                 const unsigned* __restrict__ cnt,
                                 float4* __restrict__ out4, int nNodes) {
    int gid = blockIdx.x * blockDim.x + threadIdx.x;
    int node = gid >> 4;
    int lane = gid & 15;
    if (node >= nNodes) return;

    unsigned start = ofs[node];
    unsigned n = cnt[node];

    float4 acc = make_float4(0.f, 0.f, 0.f, 0.f);
    if (n > 0) {
        int s = sorted_src[start];
        for (unsigned i = 0; i < n; ++i) {
            int s_next = (i + 1 < n) ? sorted_src[start + i + 1] : s;
            __builtin_prefetch(&x4[(size_t)s_next * 16 + lane], 0, 0);
            float4 v = x4[(size_t)s * 16 + lane];
            acc.x += v.x; acc.y += v.y; acc.z += v.z; acc.w += v.w;
            s = s_next;
        }
    }
    out4[(size_t)node * 16 + lane] = acc;
}

// ---------------------------------------------------------------------------
// Fallback scatter (only if workspace too small for CSR): native f32 atomics.
// ---------------------------------------------------------------------------
__global__ void mp_scatter_atomic_kernel(const int* __restrict__ src,
                                         const int* __restrict__ dst,
                                         const float4* __restrict__ x4,
                                         float* __restrict__ out,
                                         unsigned* __restrict__ cnt, int nE) {
    int gid = blockIdx.x * blockDim.x + threadIdx.x;
    int e = gid >> 4;
    int lane = gid & 15;
    if (e >= nE) return;

    int s = src[e];
    int d = dst[e];
    float4 v = x4[(size_t)s * 16 + lane];
    float* o = out + (size_t)d * 64 + lane * 4;
    unsafeAtomicAdd(o + 0, v.x);   // lowers to global_atomic_add_f32
    unsafeAtomicAdd(o + 1, v.y);
    unsafeAtomicAdd(o + 2, v.z);
    unsafeAtomicAdd(o + 3, v.w);
    if (lane == 0) atomicAdd(&cnt[d], 1u);
}

// ---------------------------------------------------------------------------
// Pass 5: mean division as a WMMA diagonal scale.
//   D(16x16) = diag(r)(16x16) x T(16x16),  r[m] = 1/max(cnt,1)
// decomposed into 4 chained V_WMMA_F32_16X16X4_F32 (K-chunks of 4 columns).
// One wave per 16-node x 16-feature tile; N (50000) and F (64) divide exactly,
// so EXEC is all-ones everywhere (WMMA requirement) with zero divergence.
// Layouts per CDNA5 ISA 7.12.2:
//   A 16x4 f32 : lane L holds row m=L%16; VGPR0/1 = K = (L>=16?2:0)+{0,1}
//   B 4x16 f32 : lane L holds col n=L%16; VGPR0/1 = K = (L>=16?2:0)+{0,1}
//   C/D 16x16  : VGPR j -> row j + (L>=16?8:0), col L%16
// ---------------------------------------------------------------------------
__global__ void mp_finalize_wmma_kernel(float* __restrict__ out,
                                        const unsigned* __restrict__ cnt,
                                        int nNodes) {
    int wave = blockIdx.x * 4 + (threadIdx.x >> 5);  // 4 waves per 128-thr block
    int lane = threadIdx.x & 31;
    int node_tile = wave >> 2;   // tiles along nodes
    int f_tile = wave & 3;       // 4 feature tiles (F=64)
    int n0 = node_tile * 16;
    int f0 = f_tile * 16;
    int half = lane >> 4;        // 0: lanes 0-15, 1: lanes 16-31
    int l = lane & 15;

    unsigned c = cnt[n0 + l];
    if (c < 1u) c = 1u;
    float r = 1.0f / (float)c;   // reciprocal for row m == l

    float* base = out + (size_t)n0 * 64 + f0;

    v8f acc = {};
#pragma unroll
    for (int k = 0; k < 4; ++k) {
        int kk0 = k * 4 + half * 2;  // global K index of this lane's first slot
        v2f a, b;
        a.x = (l == kk0)     ? r : 0.0f;   // A = diag(r) columns [4k, 4k+4)
        a.y = (l == kk0 + 1) ? r : 0.0f;
        b.x = base[(size_t)kk0 * 64 + l];        // B = T rows [4k, 4k+4)
        b.y = base[(size_t)(kk0 + 1) * 64 + l];
        acc = __builtin_amdgcn_wmma_f32_16x16x4_f32(
            /*neg_a=*/false, a, /*neg_b=*/false, b,
            /*c_mod=*/(short)0, acc, /*reuse_a=*/false, /*reuse_b=*/false);
    }

#pragma unroll
    for (int j = 0; j < 8; ++j) {
        base[(size_t)(j + half * 8) * 64 + l] = acc[j];
    }
}

// ---------------------------------------------------------------------------
extern "C" void kernel_launch(void* const* d_in, const int* in_sizes, int n_in,
                              void* d_out, int out_size, void* d_ws, size_t ws_size,
                              hipStream_t stream) {
    const float* x = (const float*)d_in[0];
    const int* eidx = (const int*)d_in[1];
    const int N = in_sizes[0] / 64;     // 50000
    const int E = in_sizes[1] / 2;      // 1600000
    const int* src = eidx;
    const int* dst = eidx + E;
    float* out = (float*)d_out;

    unsigned* cnt = (unsigned*)d_ws;
    size_t csr_off = (((size_t)3 * N * 4) + 255) & ~(size_t)255;
    size_t needed = csr_off + (size_t)E * 4;

    const int fin_blocks = N / 16;      // (N/16)*4 tiles, 4 waves per block

    if (ws_size >= needed) {
        // --- CSR path: ~3.2M u32 atomics instead of 102M f32 atomics ---
        unsigned* ofs = cnt + N;
        unsigned* cursor = cnt + 2 * (size_t)N;
        int* sorted_src = (int*)((char*)d_ws + csr_off);

        hipMemsetAsync(cnt, 0, (size_t)N * 4, stream);
        mp_hist_kernel<<<(E + 255) / 256, 256, 0, stream>>>(dst, cnt, E);
        mp_scan_kernel<<<1, 1024, 0, stream>>>(cnt, ofs, cursor, N);
        mp_bucket_kernel<<<(E + 255) / 256, 256, 0, stream>>>(src, dst, cursor,
                                                              sorted_src, E);
        mp_reduce_kernel<<<(N * 16 + 255) / 256, 256, 0, stream>>>(
            (const float4*)x, sorted_src, ofs, cnt, (float4*)out, N);
        mp_finalize_wmma_kernel<<<fin_blocks, 128, 0, stream>>>(out, cnt, N);
    } else {
        // --- fallback: native f32 atomic scatter ---
        hipMemsetAsync(out, 0, (size_t)N * 64 * 4, stream);
        hipMemsetAsync(cnt, 0, (size_t)N * 4, stream);
        long long threads = (long long)E * 16;
        mp_scatter_atomic_kernel<<<(unsigned)((threads + 255) / 256), 256, 0,
                                   stream>>>(src, dst, (const float4*)x, out,
                                             cnt, E);
        mp_finalize_wmma_kernel<<<fin_blocks, 128, 0, stream>>>(out, cnt, N);
    }
}